// GraphSAGE_29841432773038
// MI455X (gfx1250) — compile-verified
//
#include <hip/hip_runtime.h>
#include <stdint.h>

typedef float v2f __attribute__((ext_vector_type(2)));
typedef float v8f __attribute__((ext_vector_type(8)));

#define NN 50000
#define DFEAT 128   // d_in == d_hid == 128

// ---------------------------------------------------------------- zero
__global__ void zero_f32(float* __restrict__ p, int n) {
  int i = blockIdx.x * blockDim.x + threadIdx.x;
  if (i < n) p[i] = 0.0f;
}

// ----------------------------------------------- edge scatter (mean numerator)
// one thread per (edge, feature); D fixed at 128.
__global__ void scatter_feat(const float* __restrict__ x,
                             const int*   __restrict__ src,
                             const int*   __restrict__ dst,
                             float* __restrict__ agg,
                             float* __restrict__ cnt,   // null on 2nd layer
                             int E) {
  long long idx = (long long)blockIdx.x * blockDim.x + threadIdx.x;
  int e = (int)(idx >> 7);
  int f = (int)(idx & 127);
  if (e >= E) return;
  int s = src[e], d = dst[e];
  atomicAdd(&agg[(long long)d * DFEAT + f], x[(long long)s * DFEAT + f]);
  if (cnt != nullptr && f == 0) atomicAdd(&cnt[d], 1.0f);
}

// ----------------------------------------------- fused SAGE GEMM via fp32 WMMA
// out[m,n] = relu?( (agg[m,:]/max(cnt,1)) @ Wl.T + bias + X[m,:] @ Wr.T )
// One wave32 computes one 16x16 tile; K=128 in steps of 4 (V_WMMA_F32_16X16X4_F32).
template <int NDIM, bool RELU>
__global__ __launch_bounds__(128) void sage_gemm(
    const float* __restrict__ agg,   // [M,128] summed messages
    const float* __restrict__ cnt,   // [M] in-degree (float)
    const float* __restrict__ xin,   // [M,128] self features
    const float* __restrict__ Wl,    // [NDIM,128] row-major
    const float* __restrict__ Wr,    // [NDIM,128]
    const float* __restrict__ bias,  // [NDIM]
    float* __restrict__ out,         // [M,NDIM]
    int M) {
  constexpr int K  = 128;
  constexpr int TN = NDIM / 16;
  int wave = blockIdx.x * (blockDim.x >> 5) + (threadIdx.x >> 5);
  int tm = wave / TN;
  int tn = wave % TN;
  if (tm * 16 >= M) return;            // wave-uniform: EXEC stays all-ones

  int lane = threadIdx.x & 31;
  int half = lane >> 4;                // selects K pair within the step of 4
  int l16  = lane & 15;
  int m = tm * 16 + l16;               // A-matrix row owned by this lane
  int n = tn * 16 + l16;               // B-matrix column owned by this lane

  float inv = 1.0f / fmaxf(cnt[m], 1.0f);   // mean aggregation scale
  const float* arow  = agg + (long long)m * K;
  const float* xrow  = xin + (long long)m * K;
  const float* wlrow = Wl  + (long long)n * K;   // B[k][n] = Wl[n][k]
  const float* wrrow = Wr  + (long long)n * K;

  v8f c = {0.f, 0.f, 0.f, 0.f, 0.f, 0.f, 0.f, 0.f};

#pragma unroll
  for (int k0 = 0; k0 < K; k0 += 4) {
    int k = k0 + half * 2;
    float2 av = *(const float2*)(arow + k);
    float2 bv = *(const float2*)(wlrow + k);
    v2f a = {av.x * inv, av.y * inv};
    v2f b = {bv.x, bv.y};
    c = __builtin_amdgcn_wmma_f32_16x16x4_f32(false, a, false, b,
                                              (short)0, c, false, false);
  }
#pragma unroll
  for (int k0 = 0; k0 < K; k0 += 4) {
    int k = k0 + half * 2;
    float2 av = *(const float2*)(xrow + k);
    float2 bv = *(const float2*)(wrrow + k);
    v2f a = {av.x, av.y};
    v2f b = {bv.x, bv.y};
    c = __builtin_amdgcn_wmma_f32_16x16x4_f32(false, a, false, b,
                                              (short)0, c, false, false);
  }

  float bv = bias[n];
#pragma unroll
  for (int r = 0; r < 8; ++r) {
    int row = tm * 16 + half * 8 + r;  // C/D layout: VGPR r, halves split M by 8
    float v = c[r] + bv;
    if (RELU) v = fmaxf(v, 0.0f);
    out[(long long)row * NDIM + n] = v;
  }
}

// ----------------------------------------------- in-place log_softmax, 64 cols
__global__ void logsoftmax64(float* __restrict__ inout, int M) {
  int row = blockIdx.x * (blockDim.x >> 5) + (threadIdx.x >> 5);
  if (row >= M) return;
  int lane = threadIdx.x & 31;
  float* p = inout + (long long)row * 64;
  float v0 = p[lane];
  float v1 = p[lane + 32];
  float mx = fmaxf(v0, v1);
#pragma unroll
  for (int off = 16; off > 0; off >>= 1) mx = fmaxf(mx, __shfl_xor(mx, off, 32));
  float s = expf(v0 - mx) + expf(v1 - mx);
#pragma unroll
  for (int off = 16; off > 0; off >>= 1) s += __shfl_xor(s, off, 32);
  float lse = logf(s) + mx;
  p[lane]      = v0 - lse;
  p[lane + 32] = v1 - lse;
}

// ---------------------------------------------------------------- launcher
extern "C" void kernel_launch(void* const* d_in, const int* in_sizes, int n_in,
                              void* d_out, int out_size, void* d_ws, size_t ws_size,
                              hipStream_t stream) {
  const float* x    = (const float*)d_in[0];
  const int*   eidx = (const int*)d_in[1];     // [2,E] flattened
  const float* W1l  = (const float*)d_in[2];
  const float* b1   = (const float*)d_in[3];
  const float* W1r  = (const float*)d_in[4];
  const float* W2l  = (const float*)d_in[5];
  const float* b2   = (const float*)d_in[6];
  const float* W2r  = (const float*)d_in[7];

  const int E = in_sizes[1] / 2;
  const int* src = eidx;
  const int* dst = eidx + E;

  // workspace: agg[NN*128] | cnt[NN] | h[NN*128]  (fp32)
  float* agg = (float*)d_ws;
  float* cnt = agg + (long long)NN * DFEAT;
  float* h   = cnt + NN;
  float* outp = (float*)d_out;

  const int zn1 = NN * DFEAT + NN;            // agg + cnt contiguous
  zero_f32<<<(zn1 + 255) / 256, 256, 0, stream>>>(agg, zn1);

  {
    long long threads = (long long)E * DFEAT;
    int blocks = (int)((threads + 255) / 256);
    scatter_feat<<<blocks, 256, 0, stream>>>(x, src, dst, agg, cnt, E);
  }

  {
    // 3125 M-tiles * 8 N-tiles = 25000 waves, 4 waves/block
    int waves = (NN / 16) * (DFEAT / 16);
    sage_gemm<DFEAT, true><<<waves / 4, 128, 0, stream>>>(
        agg, cnt, x, W1l, W1r, b1, h, NN);
  }

  zero_f32<<<(NN * DFEAT + 255) / 256, 256, 0, stream>>>(agg, NN * DFEAT);

  {
    long long threads = (long long)E * DFEAT;
    int blocks = (int)((threads + 255) / 256);
    scatter_feat<<<blocks, 256, 0, stream>>>(h, src, dst, agg, nullptr, E);
  }

  {
    // 3125 M-tiles * 4 N-tiles = 12500 waves
    int waves = (NN / 16) * (64 / 16);
    sage_gemm<64, false><<<waves / 4, 128, 0, stream>>>(
        agg, cnt, h, W2l, W2r, b2, outp, NN);
  }

  {
    int waves_per_block = 256 / 32;
    int blocks = (NN + waves_per_block - 1) / waves_per_block;
    logsoftmax64<<<blocks, 256, 0, stream>>>(outp, NN);
  }
}